// QuantumEncoder_1726576853345
// MI455X (gfx1250) — compile-verified
//
#include <hip/hip_runtime.h>

typedef __attribute__((ext_vector_type(2))) float v2f;
typedef __attribute__((ext_vector_type(4))) float f4;
typedef __attribute__((ext_vector_type(8))) float v8f;

#define NQ       14
#define DSTATE   16384
#define NROWS    8192      // BATCH * SEQ
#define ROWTILES 512       // NROWS / 16
#define COLTILES 1024      // DSTATE / 16
#define WPB      8         // waves per block in main kernel

// ---------------------------------------------------------------------------
// Kernel 0: stream 1 GiB of zeros to d_out with non-temporal B128 stores.
// This is the roofline floor (~46us @ 23.3 TB/s); NT keeps W2 resident in L2.
// ---------------------------------------------------------------------------
__global__ void zero_out_kernel(f4* __restrict__ out, long n4) {
    long i      = (long)blockIdx.x * blockDim.x + threadIdx.x;
    long stride = (long)gridDim.x * blockDim.x;
    f4 z = {0.0f, 0.0f, 0.0f, 0.0f};
    for (; i < n4; i += stride)
        __builtin_nontemporal_store(z, out + i);
}

// ---------------------------------------------------------------------------
// Kernel 1: the scalar complex "total" = prod(rot_factors) * prod(ent_factors)
// (independent of x). One lane; negligible cost. ws[0]=re, ws[1]=im.
// ---------------------------------------------------------------------------
__global__ void total_kernel(const float* __restrict__ rp,
                             const float* __restrict__ ep,
                             float* __restrict__ ws) {
    if (threadIdx.x != 0 || blockIdx.x != 0) return;
    float re = 1.0f, im = 0.0f;
    for (int q = 0; q < NQ; ++q) {
        float h0 = 0.5f * rp[q * 3 + 0];
        float h1 = 0.5f * rp[q * 3 + 1];
        float h2 = 0.5f * rp[q * 3 + 2];
        float fr = cosf(h0) * cosf(h1) * cosf(h2);
        float fi = sinf(h0) * sinf(h1) * sinf(h2);
        float nre = re * fr - im * fi;
        float nim = re * fi + im * fr;
        re = nre; im = nim;
    }
    for (int p = 0; p < NQ - 1; ++p) {
        float cs = 1.0f / (1.0f + __expf(-ep[p]));
        float fr = cs, fi = 1.0f - cs;
        float nre = re * fr - im * fi;
        float nim = re * fi + im * fr;
        re = nre; im = nim;
    }
    ws[0] = re; ws[1] = im;
}

// ---------------------------------------------------------------------------
// Kernel 2: H[i][q] = tanh(x[i]*W1[q] + b1[q]), K-padded to 16 with zeros.
// H is the WMMA A-matrix source: [8192 x 16] f32 in workspace.
// ---------------------------------------------------------------------------
__global__ void encode_kernel(const float* __restrict__ x,
                              const float* __restrict__ W1,
                              const float* __restrict__ b1,
                              float* __restrict__ H) {
    int i = blockIdx.x * blockDim.x + threadIdx.x;
    if (i >= NROWS) return;
    float xv = x[i];
#pragma unroll
    for (int q = 0; q < 16; ++q) {
        float v = 0.0f;
        if (q < NQ) v = tanhf(xv * W1[q] + b1[q]);
        H[i * 16 + q] = v;
    }
}

// ---------------------------------------------------------------------------
// Kernel 3: per 16-row tile, L = H_tile @ W2 (+b2) via 4x chained
// V_WMMA_F32_16X16X4_F32 per 16-col tile; online sum of exp(L) per row
// (logits are O(1) -> max-subtraction-free softmax is exact enough);
// column-0 logit captured from tile j=0; final w0 = exp(l0)/sum scaled by
// the complex "total" and scattered to out[row, 0].
// C/D layout: row = v + 8*(lane/16), col = lane%16  -> row-sum reduces over
// the 16 lanes of each half (shfl_xor 1,2,4,8).
// ---------------------------------------------------------------------------
__global__ __launch_bounds__(32 * WPB)
void softmax0_wmma_kernel(const float* __restrict__ H,
                          const float* __restrict__ W2,
                          const float* __restrict__ b2,
                          const float* __restrict__ totalv,
                          float* __restrict__ out) {
    __shared__ float s_partial[WPB][16];
    __shared__ float s_l0[16];

    const int tile  = blockIdx.x;          // 0..511
    const int lane  = threadIdx.x & 31;
    const int wave  = threadIdx.x >> 5;
    const int m     = lane & 15;           // column within tile / row within tile
    const int khalf = lane >> 4;           // 0: K pair {0,1}; 1: K pair {2,3}
    const int row0  = tile * 16;

    // A fragments (wave-invariant across the column loop).
    // f32 16x4 A layout: lane m holds row m; VGPR0=K(2*khalf), VGPR1=K(2*khalf+1).
    v2f a[4];
#pragma unroll
    for (int kb = 0; kb < 4; ++kb) {
        const float* hp = H + (long)(row0 + m) * 16 + kb * 4 + 2 * khalf;
        a[kb].x = hp[0];
        a[kb].y = hp[1];
    }

    float sacc[8] = {0, 0, 0, 0, 0, 0, 0, 0};
    float l0[8]   = {0, 0, 0, 0, 0, 0, 0, 0};

    for (int j = wave; j < COLTILES; j += WPB) {
        const int d0  = j << 4;
        const float bv = b2[d0 + m];
        v8f c = {bv, bv, bv, bv, bv, bv, bv, bv};
#pragma unroll
        for (int kb = 0; kb < 4; ++kb) {
            const int k0 = kb * 4 + 2 * khalf;
            // Branchless zero-pad of W2 rows 14,15: clamped address + select
            // (no EXEC change -> WMMA EXEC==all-ones requirement preserved).
            const int r0 = (k0     < NQ) ? k0     : (NQ - 1);
            const int r1 = (k0 + 1 < NQ) ? k0 + 1 : (NQ - 1);
            float w0v = W2[(long)r0 * DSTATE + d0 + m];
            float w1v = W2[(long)r1 * DSTATE + d0 + m];
            v2f b;
            b.x = (k0     < NQ) ? w0v : 0.0f;
            b.y = (k0 + 1 < NQ) ? w1v : 0.0f;
            c = __builtin_amdgcn_wmma_f32_16x16x4_f32(
                    /*neg_a=*/false, a[kb], /*neg_b=*/false, b,
                    /*c_mod=*/(short)0, c, /*reuse_a=*/false, /*reuse_b=*/false);
        }
        if (j == 0) {            // only wave 0 ever hits this; captures d==0 logits
#pragma unroll
            for (int v = 0; v < 8; ++v) l0[v] = c[v];
        }
#pragma unroll
        for (int v = 0; v < 8; ++v) sacc[v] += __expf(c[v]);
    }

    // Reduce each row's partial sum across the 16 lanes of its half.
#pragma unroll
    for (int v = 0; v < 8; ++v) {
        float s = sacc[v];
        s += __shfl_xor(s, 1, 32);
        s += __shfl_xor(s, 2, 32);
        s += __shfl_xor(s, 4, 32);
        s += __shfl_xor(s, 8, 32);
        sacc[v] = s;
    }

    if (m == 0) {  // lanes 0 and 16 hold rows v and v+8 respectively
#pragma unroll
        for (int v = 0; v < 8; ++v)
            s_partial[wave][v + 8 * khalf] = sacc[v];
        if (wave == 0) {
#pragma unroll
            for (int v = 0; v < 8; ++v)
                s_l0[v + 8 * khalf] = l0[v];   // logit at d==0 (col N=0)
        }
    }
    __syncthreads();

    if (threadIdx.x < 16) {
        const int r = threadIdx.x;
        float denom = 0.0f;
#pragma unroll
        for (int w = 0; w < WPB; ++w) denom += s_partial[w][r];
        const float w0  = __expf(s_l0[r]) / denom;
        const float tre = totalv[0];
        const float tim = totalv[1];
        const long base = (long)(row0 + r) * (DSTATE * 2);  // complex64 as 2 floats
        out[base]     = tre * w0;
        out[base + 1] = tim * w0;
    }
}

// ---------------------------------------------------------------------------
extern "C" void kernel_launch(void* const* d_in, const int* in_sizes, int n_in,
                              void* d_out, int out_size, void* d_ws, size_t ws_size,
                              hipStream_t stream) {
    const float* x  = (const float*)d_in[0];  // [4,2048]
    const float* rp = (const float*)d_in[1];  // [14,3]
    const float* ep = (const float*)d_in[2];  // [13]
    const float* W1 = (const float*)d_in[3];  // [1,14]
    const float* b1 = (const float*)d_in[4];  // [14]
    const float* W2 = (const float*)d_in[5];  // [14,16384]
    const float* b2 = (const float*)d_in[6];  // [16384]

    float* out   = (float*)d_out;
    float* ws    = (float*)d_ws;
    float* total = ws;        // 2 floats (padded to 16)
    float* H     = ws + 16;   // 8192 x 16 floats = 512 KB

    // 1 GiB of zeros: 8192*16384*2 floats -> 67,108,864 f4 stores.
    const long n4 = (long)NROWS * DSTATE * 2 / 4;
    zero_out_kernel<<<8192, 256, 0, stream>>>((f4*)d_out, n4);

    total_kernel<<<1, 32, 0, stream>>>(rp, ep, total);
    encode_kernel<<<NROWS / 256, 256, 0, stream>>>(x, W1, b1, H);
    softmax0_wmma_kernel<<<ROWTILES, 32 * WPB, 0, stream>>>(H, W2, b2, total, out);
    (void)in_sizes; (void)n_in; (void)out_size; (void)ws_size;
}